// FilterLegalMoves_16475494548159
// MI455X (gfx1250) — compile-verified
//
#include <hip/hip_runtime.h>
#include <stdint.h>

// FilterLegalMoves: out[b,a] = (a in possible_moves[b,:] && x[b,a] != 0) ? x[b,a] : -1e9
// B=4096, A=30800, K=1024. Pure memory-bound scatter/compose -> no WMMA.
// Strategy: one workgroup per row, compose the whole row in LDS (123KB -- needs
// CDNA5's 320KB/WGP), then stream it out with wide non-temporal stores.

#define A_DIM   30800
#define A_VEC   (A_DIM / 4)   // 7700 float4 per row
#define K_DIM   1024
#define THREADS 1024
#define NEG_INF (-1000000000.0f)

typedef float v4f __attribute__((ext_vector_type(4)));

__global__ __launch_bounds__(THREADS)
void FilterLegalMoves_16475494548159_kernel(const float* __restrict__ x,
                                            const int*   __restrict__ moves,
                                            float*       __restrict__ out) {
  __shared__ v4f row4[A_VEC];    // 123,200 B: the composed output row
  __shared__ int idxbuf[K_DIM];  //   4,096 B: this row's legal-move indices
  float* row = (float*)row4;

  const int b = blockIdx.x;
  const int t = threadIdx.x;

  // ---- Stage 1: async DMA the row's K int32 indices global -> LDS ---------
  // (GVS form: vdst = LDS byte addr, vaddr = 32-bit offset, saddr = 64b base)
  // Overlaps with the LDS fill below; tracked by ASYNCcnt.
  {
    uint64_t sbase    = (uint64_t)(uintptr_t)(moves + (size_t)b * K_DIM);
    unsigned voff     = (unsigned)t * 4u;
    unsigned lds_addr = (unsigned)(uintptr_t)(&idxbuf[t]);
    asm volatile("global_load_async_to_lds_b32 %0, %1, %2"
                 :: "v"(lds_addr), "v"(voff), "s"(sbase)
                 : "memory");
  }

  // ---- Stage 2: fill the LDS row with -1e9 (ds_store_b128) -----------------
  v4f ninf = {NEG_INF, NEG_INF, NEG_INF, NEG_INF};
  for (int i = t; i < A_VEC; i += THREADS)
    row4[i] = ninf;

  asm volatile("s_wait_asynccnt 0" ::: "memory");
  __syncthreads();

  // ---- Stage 3: gather the legal logits, scatter into the LDS row ----------
  // Duplicate indices write identical values -> benign race.
  const float* xrow = x + (size_t)b * A_DIM;
  for (int k = t; k < K_DIM; k += THREADS) {
    int   idx = idxbuf[k];
    float v   = xrow[idx];
    row[idx]  = (v == 0.0f) ? NEG_INF : v;   // matches reference masked==0 rule
  }
  __syncthreads();

  // ---- Stage 4: stream the finished row out, non-temporal b128 stores ------
  // (504 MB total output; NT hint keeps it from thrashing the 192MB L2.)
  v4f* orow = (v4f*)(out + (size_t)b * A_DIM);
  for (int i = t; i < A_VEC; i += THREADS)
    __builtin_nontemporal_store(row4[i], orow + i);
}

extern "C" void kernel_launch(void* const* d_in, const int* in_sizes, int n_in,
                              void* d_out, int out_size, void* d_ws, size_t ws_size,
                              hipStream_t stream) {
  const float* x     = (const float*)d_in[0];
  const int*   moves = (const int*)d_in[1];
  float*       out   = (float*)d_out;

  const int Bn = in_sizes[0] / A_DIM;   // 4096
  hipLaunchKernelGGL(FilterLegalMoves_16475494548159_kernel,
                     dim3(Bn), dim3(THREADS), 0, stream,
                     x, moves, out);
  (void)n_in; (void)out_size; (void)d_ws; (void)ws_size;
}